// LGN_SD_66142496359035
// MI455X (gfx1250) — compile-verified
//
#include <hip/hip_runtime.h>
#include <math.h>

#define NFREQ 25
#define SCAN_THREADS 256

typedef __attribute__((ext_vector_type(2))) float v2f;
typedef __attribute__((ext_vector_type(8))) float v8f;

// ---------- 2x2 matrix helpers ----------
struct M2 { float a, b, c, d; };   // [[a,b],[c,d]]

__device__ __forceinline__ M2 mul2(const M2& X, const M2& Y) {
    M2 r;
    r.a = fmaf(X.a, Y.a, X.b * Y.c);
    r.b = fmaf(X.a, Y.b, X.b * Y.d);
    r.c = fmaf(X.c, Y.a, X.d * Y.c);
    r.d = fmaf(X.c, Y.b, X.d * Y.d);
    return r;
}

__device__ __forceinline__ float softplus_f(float x) {
    return (x > 20.0f) ? x : log1pf(expf(x));
}

// A(t) = [[-d0, s], [-s, -d1]] from Fourier-feature MLP
__device__ __forceinline__ M2 getA(float tt,
                                   const float* __restrict__ freqs,
                                   const float* __restrict__ Sw, float sb,
                                   const float* __restrict__ Dw,
                                   float db0, float db1) {
    float s = sb, d0 = db0, d1 = db1;
#pragma unroll
    for (int j = 0; j < NFREQ; ++j) {
        float ph = tt * freqs[j];
        float sn, cs;
        sincosf(ph, &sn, &cs);
        // features: f[j] = cos, f[j+25] = sin  (Sw: (1,50); Dw: (2,50) row-major)
        s  = fmaf(cs, Sw[j],               s );
        s  = fmaf(sn, Sw[j + NFREQ],       s );
        d0 = fmaf(cs, Dw[j],               d0);
        d0 = fmaf(sn, Dw[j + NFREQ],       d0);
        d1 = fmaf(cs, Dw[2*NFREQ + j],         d1);
        d1 = fmaf(sn, Dw[2*NFREQ + NFREQ + j], d1);
    }
    d0 = softplus_f(d0);
    d1 = softplus_f(d1);
    M2 A;
    A.a = -d0; A.b =  s;
    A.c = -s;  A.d = -d1;
    return A;
}

// exact closed-form expm for 2x2
__device__ __forceinline__ M2 expm2(const M2& m) {
    float mu = 0.5f * (m.a + m.d);
    float p  = 0.5f * (m.a - m.d);
    float q  = fmaf(p, p, m.b * m.c);
    float C, S;
    const float eps = 1e-12f;
    if (q > eps) {
        float w  = sqrtf(q);
        float ew = expf(w);
        float iw = 1.0f / ew;
        C = 0.5f * (ew + iw);
        S = 0.5f * (ew - iw) / w;
    } else if (q < -eps) {
        float w = sqrtf(-q);
        C = cosf(w);
        S = sinf(w) / w;
    } else {
        C = 1.0f + 0.5f * q;
        S = 1.0f + q * (1.0f / 6.0f);
    }
    float e = expf(mu);
    M2 r;
    r.a = e * fmaf(S,  p, C);
    r.b = e * (S * m.b);
    r.c = e * (S * m.c);
    r.d = e * fmaf(S, -p, C);
    return r;
}

// ---------- kernel 1: per-interval Magnus-2 propagator E_i ----------
__global__ void k_compute_E(const float* __restrict__ t,
                            const float* __restrict__ freqs,
                            const float* __restrict__ Sw,
                            const float* __restrict__ Sb,
                            const float* __restrict__ Dw,
                            const float* __restrict__ Db,
                            float* __restrict__ Ef, int Tm1) {
    int i = blockIdx.x * blockDim.x + threadIdx.x;
    if (i >= Tm1) return;
    float sb  = Sb[0];
    float db0 = Db[0], db1 = Db[1];
    float t0 = t[i], t1 = t[i + 1];
    float dt = t1 - t0;
    float tm = fmaf(0.5f, dt, t0);

    M2 A0 = getA(t0, freqs, Sw, sb, Dw, db0, db1);
    M2 Am = getA(tm, freqs, Sw, sb, Dw, db0, db1);
    M2 A1 = getA(t1, freqs, Sw, sb, Dw, db0, db1);

    M2 c1 = mul2(A0, A1);
    M2 c2 = mul2(A1, A0);
    float k = dt * dt * (1.0f / 12.0f);
    M2 Om;
    Om.a = fmaf(Am.a, dt, k * (c1.a - c2.a));
    Om.b = fmaf(Am.b, dt, k * (c1.b - c2.b));
    Om.c = fmaf(Am.c, dt, k * (c1.c - c2.c));
    Om.d = fmaf(Am.d, dt, k * (c1.d - c2.d));

    M2 E = expm2(Om);
    Ef[i * 4 + 0] = E.a;
    Ef[i * 4 + 1] = E.b;
    Ef[i * 4 + 2] = E.c;
    Ef[i * 4 + 3] = E.d;
}

// ---------- kernel 2: single-block parallel prefix product ----------
// V_0 = I, V_j = E_{j-1}; inclusive left-scan: P_j = V_j @ ... @ V_0
__global__ void k_scan_P(const float* __restrict__ Ef,
                         float* __restrict__ Pf, int T) {
    __shared__ float4 lds[SCAN_THREADS];
    const int tid = threadIdx.x;
    const int seg = (T + SCAN_THREADS - 1) / SCAN_THREADS;
    const int jb  = tid * seg;
    const int je  = (jb + seg < T) ? (jb + seg) : T;

    M2 L = {1.f, 0.f, 0.f, 1.f};
    for (int j = jb; j < je; ++j) {
        M2 V;
        if (j == 0) { V = {1.f, 0.f, 0.f, 1.f}; }
        else {
            V.a = Ef[(j - 1) * 4 + 0];
            V.b = Ef[(j - 1) * 4 + 1];
            V.c = Ef[(j - 1) * 4 + 2];
            V.d = Ef[(j - 1) * 4 + 3];
        }
        L = mul2(V, L);
    }
    lds[tid] = make_float4(L.a, L.b, L.c, L.d);
    __syncthreads();

    for (int o = 1; o < SCAN_THREADS; o <<= 1) {
        float4 mine = lds[tid];
        float4 prevv = mine;
        bool has = (tid >= o);
        if (has) prevv = lds[tid - o];
        __syncthreads();
        if (has) {
            M2 X = {mine.x, mine.y, mine.z, mine.w};
            M2 Y = {prevv.x, prevv.y, prevv.z, prevv.w};
            M2 Z = mul2(X, Y);
            lds[tid] = make_float4(Z.a, Z.b, Z.c, Z.d);
        }
        __syncthreads();
    }

    M2 acc = {1.f, 0.f, 0.f, 1.f};
    if (tid > 0) {
        float4 pp = lds[tid - 1];
        acc = {pp.x, pp.y, pp.z, pp.w};
    }
    for (int j = jb; j < je; ++j) {
        M2 V;
        if (j == 0) { V = {1.f, 0.f, 0.f, 1.f}; }
        else {
            V.a = Ef[(j - 1) * 4 + 0];
            V.b = Ef[(j - 1) * 4 + 1];
            V.c = Ef[(j - 1) * 4 + 2];
            V.d = Ef[(j - 1) * 4 + 3];
        }
        acc = mul2(V, acc);
        Pf[j * 4 + 0] = acc.a;
        Pf[j * 4 + 1] = acc.b;
        Pf[j * 4 + 2] = acc.c;
        Pf[j * 4 + 3] = acc.d;
    }
}

// ---------- kernel 3: out(2T x B) = P(2T x 2) @ x0(2 x B) via WMMA f32 16x16x4 ----------
// Compile-time stride BC so per-row store offsets fold into the 24-bit IOFFSET.
// Each wave: one 16-row M tile, NT=4 consecutive 16-col N tiles (A reused, 4 unrolled WMMAs).
// A layout (ISA 16x4 f32): lanes 0-15 hold {K=0,K=1} for M=lane; lanes 16-31 hold {K=2,K=3}=0
// B layout (4x16 f32, mirrored): lanes 0-15 hold {K=0,K=1} for N=lane; lanes 16-31 = 0
// D layout: VGPR v -> row v (lanes 0-15) / row v+8 (lanes 16-31), col = lane%16
template<int BC>
__global__ void k_apply_P_c(const float* __restrict__ Pf,
                            const float* __restrict__ x0,
                            float* __restrict__ out) {
    constexpr int NT = 4;
    const int lane = threadIdx.x & 31;
    const int wave = threadIdx.x >> 5;
    const int row0 = blockIdx.y * 16;
    const int col0 = (blockIdx.x * (blockDim.x >> 5) + wave) * (16 * NT);
    const bool lo  = lane < 16;
    const int  l15 = lane & 15;

    v2f a;
    if (lo) {
        const int row = row0 + lane;
        a.x = Pf[row * 2 + 0];
        a.y = Pf[row * 2 + 1];
    } else {
        a.x = 0.f; a.y = 0.f;
    }

    const int rbase = row0 + (lo ? 0 : 8);

#pragma unroll
    for (int nt = 0; nt < NT; ++nt) {
        const int ctile = col0 + nt * 16;
        v2f b;
        if (lo) {
            const int col = ctile + lane;
            b.x = x0[col];
            b.y = x0[BC + col];
        } else {
            b.x = 0.f; b.y = 0.f;
        }
        v8f c = {};
        v8f d = __builtin_amdgcn_wmma_f32_16x16x4_f32(
            false, a, false, b, (short)0, c, false, false);

        float* __restrict__ p = out + (size_t)rbase * (size_t)BC + (size_t)(ctile + l15);
#pragma unroll
        for (int v = 0; v < 8; ++v) {
            p[v * BC] = d[v];   // immediate offset: v*BC*4 bytes (fits 24-bit IOFFSET)
        }
    }
}

// Generic-B fallback (runtime stride)
__global__ void k_apply_P_g(const float* __restrict__ Pf,
                            const float* __restrict__ x0,
                            float* __restrict__ out, int B) {
    const int lane = threadIdx.x & 31;
    const int wave = threadIdx.x >> 5;
    const int row0 = blockIdx.y * 16;
    const int col0 = (blockIdx.x * (blockDim.x >> 5) + wave) * 16;
    const bool lo  = lane < 16;
    const int  l15 = lane & 15;

    v2f a, b;
    if (lo) {
        const int row = row0 + lane;
        a.x = Pf[row * 2 + 0];
        a.y = Pf[row * 2 + 1];
        const int col = col0 + lane;
        b.x = x0[col];
        b.y = x0[B + col];
    } else {
        a.x = 0.f; a.y = 0.f;
        b.x = 0.f; b.y = 0.f;
    }
    v8f c = {};
    v8f d = __builtin_amdgcn_wmma_f32_16x16x4_f32(
        false, a, false, b, (short)0, c, false, false);

    const int colw  = col0 + l15;
    const int rbase = row0 + (lo ? 0 : 8);
#pragma unroll
    for (int v = 0; v < 8; ++v) {
        out[(size_t)(rbase + v) * (size_t)B + (size_t)colw] = d[v];
    }
}

extern "C" void kernel_launch(void* const* d_in, const int* in_sizes, int n_in,
                              void* d_out, int out_size, void* d_ws, size_t ws_size,
                              hipStream_t stream) {
    const float* t     = (const float*)d_in[0];
    const float* x0    = (const float*)d_in[1];
    const float* freqs = (const float*)d_in[2];
    const float* Sw    = (const float*)d_in[3];
    const float* Sb    = (const float*)d_in[4];
    const float* Dw    = (const float*)d_in[5];
    const float* Db    = (const float*)d_in[6];
    float* out = (float*)d_out;

    const int T = in_sizes[0];         // 4096
    const int B = in_sizes[1] / 2;     // 8192
    const int Tm1 = T - 1;

    float* Ef = (float*)d_ws;          // (T-1)*4 floats
    float* Pf = Ef + (size_t)T * 4;    // T*4 floats

    k_compute_E<<<(Tm1 + 255) / 256, 256, 0, stream>>>(t, freqs, Sw, Sb, Dw, Db, Ef, Tm1);
    k_scan_P<<<1, SCAN_THREADS, 0, stream>>>(Ef, Pf, T);

    if (B == 8192) {
        // 512 N tiles / (4 per wave * 8 waves) = 16 blocks in x; 512 M tiles in y
        dim3 grid(B / 16 / 4 / 8, (2 * T) / 16);
        k_apply_P_c<8192><<<grid, 256, 0, stream>>>(Pf, x0, out);
    } else {
        dim3 grid(B / 16 / 8, (2 * T) / 16);
        k_apply_P_g<<<grid, 256, 0, stream>>>(Pf, x0, out, B);
    }
}